// TriMulBmmOutputWrapper_83708912599798
// MI455X (gfx1250) — compile-verified
//
#include <hip/hip_runtime.h>
#include <math.h>

typedef float v2f __attribute__((ext_vector_type(2)));
typedef float v8f __attribute__((ext_vector_type(8)));

#define NN 512
#define DD 128
#define NPIX (NN * NN)

// ---------------------------------------------------------------------------
// CDNA5 async copy: global -> LDS, 16 bytes per lane, tracked by ASYNCcnt.
// vdst = per-lane LDS byte address (low 32 bits of the generic LDS pointer),
// vaddr = per-lane 64-bit global address.
// ---------------------------------------------------------------------------
__device__ __forceinline__ void async_ld16(void* lds_dst, const void* gsrc) {
  const unsigned loff = (unsigned)(unsigned long long)(uintptr_t)lds_dst;
  asm volatile("global_load_async_to_lds_b128 %0, %1, off"
               :
               : "v"(loff), "v"(gsrc)
               : "memory");
}

// ---------------------------------------------------------------------------
// Kernel 1: batched per-channel GEMM  P_d = A_d * B_d  (f32, WMMA 16x16x4)
// Grid: (N/128, N/128, D), block = 256 threads (8 waves).
// Double-buffered LDS panels filled with GLOBAL_LOAD_ASYNC_TO_LDS_B128;
// each wave owns 64x32 (4x2 tiles of 16x16 => 8 v8f accumulators).
// ---------------------------------------------------------------------------
__global__ __launch_bounds__(256) void k_bmm(const float* __restrict__ A_,
                                             const float* __restrict__ B_,
                                             float* __restrict__ P_) {
  const int d  = blockIdx.z;
  const int ti = blockIdx.y;
  const int tj = blockIdx.x;
  const float* A = A_ + (size_t)d * NN * NN;
  const float* B = B_ + (size_t)d * NN * NN;
  float*       P = P_ + (size_t)d * NN * NN;

  // A stride 20 floats (80 B: 16B-aligned rows; banks (20r+k)%64 distinct for
  // the 16 fragment rows; 20x==2 (mod 64) unsolvable -> no half-wave clash).
  // B stride 136 floats (544 B: 16B-aligned; 2*136%64==16 -> the two half-wave
  // K rows use disjoint bank sets).
  __shared__ __align__(16) float As[2][128 * 20];
  __shared__ __align__(16) float Bs[2][16 * 136];

  const int tid   = threadIdx.x;
  const int wave  = tid >> 5;
  const int lane  = tid & 31;
  const int lrow  = lane & 15;
  const int lhalf = lane >> 4;   // 0: K = k0,k0+1   1: K = k0+2,k0+3
  const int wrow  = wave >> 2;   // 0..1  -> 64-row half
  const int wcol  = wave & 3;    // 0..3  -> 32-col quarter

  const int i0 = ti * 128;
  const int j0 = tj * 128;

  // per-thread async-copy slots (2 x A, 2 x B float4 per chunk)
  int aAr[2], aAc[2], aBr[2], aBc[2];
  for (int e = 0; e < 2; ++e) {
    const int idx = e * 256 + tid;   // 0..511
    aAr[e] = idx >> 2;               // A row 0..127
    aAc[e] = (idx & 3) * 4;          // A k   0/4/8/12
    aBr[e] = idx >> 5;               // B k   0..15
    aBc[e] = (idx & 31) * 4;         // B col 0..124
  }

  const v8f vzero = {0.f, 0.f, 0.f, 0.f, 0.f, 0.f, 0.f, 0.f};
  v8f acc[4][2];
  for (int r = 0; r < 4; ++r)
    for (int c = 0; c < 2; ++c)
      acc[r][c] = vzero;

  // prologue: async-fill buffer 0 with chunk 0
  for (int e = 0; e < 2; ++e)
    async_ld16(&As[0][aAr[e] * 20 + aAc[e]],
               A + (size_t)(i0 + aAr[e]) * NN + aAc[e]);
  for (int e = 0; e < 2; ++e)
    async_ld16(&Bs[0][aBr[e] * 136 + aBc[e]],
               B + (size_t)aBr[e] * NN + j0 + aBc[e]);

  for (int kc = 0; kc < NN / 16; ++kc) {
    const int cur = kc & 1;
    if (kc + 1 < NN / 16) {
      const int nxt = cur ^ 1;
      const int kn  = (kc + 1) * 16;
      for (int e = 0; e < 2; ++e)
        async_ld16(&As[nxt][aAr[e] * 20 + aAc[e]],
                   A + (size_t)(i0 + aAr[e]) * NN + kn + aAc[e]);
      for (int e = 0; e < 2; ++e)
        async_ld16(&Bs[nxt][aBr[e] * 136 + aBc[e]],
                   B + (size_t)(kn + aBr[e]) * NN + j0 + aBc[e]);
      // 8 outstanding; async loads complete in order -> <=4 means chunk kc landed
      asm volatile("s_wait_asynccnt 0x4" ::: "memory");
    } else {
      asm volatile("s_wait_asynccnt 0x0" ::: "memory");
    }
    __syncthreads();

    for (int ks = 0; ks < 16; ks += 4) {
      const int kk = ks + 2 * lhalf;
      v2f af[4];
      for (int rt = 0; rt < 4; ++rt) {
        const int row = wrow * 64 + rt * 16 + lrow;
        af[rt].x = As[cur][row * 20 + kk];
        af[rt].y = As[cur][row * 20 + kk + 1];
      }
      v2f bf[2];
      for (int ct = 0; ct < 2; ++ct) {
        const int col = wcol * 32 + ct * 16 + lrow;
        bf[ct].x = Bs[cur][kk * 136 + col];
        bf[ct].y = Bs[cur][(kk + 1) * 136 + col];
      }
      for (int rt = 0; rt < 4; ++rt)
        for (int ct = 0; ct < 2; ++ct)
          acc[rt][ct] = __builtin_amdgcn_wmma_f32_16x16x4_f32(
              false, af[rt], false, bf[ct], (short)0, acc[rt][ct], false, false);
    }
    __syncthreads();   // all waves done with 'cur' before it is refilled
  }

  // store: C/D layout -> VGPR v holds row (v + 8*lhalf), col = lane&15
  for (int rt = 0; rt < 4; ++rt) {
    for (int ct = 0; ct < 2; ++ct) {
      const int colg = j0 + wcol * 32 + ct * 16 + lrow;
      for (int v = 0; v < 8; ++v) {
        const int rowg = i0 + wrow * 64 + rt * 16 + v + 8 * lhalf;
        P[(size_t)rowg * NN + colg] = acc[rt][ct][v];
      }
    }
  }
}

// ---------------------------------------------------------------------------
// Kernel 2: per-pixel LayerNorm(D) -> linear_z (WMMA) and sigmoid(linear_g)
// gate (WMMA), fused.  Persistent workgroups; weights transposed into LDS
// once per workgroup.  Tile = 64 pixels x 128 channels.
// ---------------------------------------------------------------------------
__global__ __launch_bounds__(256) void k_fused(const float* __restrict__ p,
                                               const float* __restrict__ zn,
                                               const float* __restrict__ lnw,
                                               const float* __restrict__ lnb,
                                               const float* __restrict__ Wz,
                                               const float* __restrict__ bz,
                                               const float* __restrict__ Wg,
                                               const float* __restrict__ bg,
                                               float* __restrict__ out) {
  extern __shared__ float smem[];
  float* BzT = smem;                  // [128 d][136] = Wz^T, B matrix for Z
  float* BgT = BzT + 128 * 136;       // [128 o][136] = Wg^T, B matrix for G
  float* tP  = BgT + 128 * 136;       // [64 pix][132] p tile (then normalized)
  float* tZ  = tP + 64 * 132;         // [64 pix][132] z_norm tile

  __shared__ float lnwS[128], lnbS[128], bzS[128], bgS[128];
  __shared__ float redS[256], redQ[256], muA[64], rsA[64];

  const int tid   = threadIdx.x;
  const int wave  = tid >> 5;
  const int lane  = tid & 31;
  const int lrow  = lane & 15;
  const int lhalf = lane >> 4;
  const int wrow  = wave >> 1;   // 0..3 : 16-row tile
  const int wch   = wave & 1;    // 0..1 : 64-col half

  // Load both weight matrices transposed into LDS (coalesced reads).
  for (int e = 0; e < 16; ++e) {
    const int idx = e * 256 + tid;   // 0..4095 float4 slots
    const int c   = idx >> 5;        // out-channel row of W
    const int d4  = (idx & 31) * 4;  // in-channel
    const float4 vz = *(const float4*)(Wz + (size_t)c * 128 + d4);
    const float4 vg = *(const float4*)(Wg + (size_t)c * 128 + d4);
    BzT[(d4 + 0) * 136 + c] = vz.x;
    BzT[(d4 + 1) * 136 + c] = vz.y;
    BzT[(d4 + 2) * 136 + c] = vz.z;
    BzT[(d4 + 3) * 136 + c] = vz.w;
    BgT[(d4 + 0) * 136 + c] = vg.x;
    BgT[(d4 + 1) * 136 + c] = vg.y;
    BgT[(d4 + 2) * 136 + c] = vg.z;
    BgT[(d4 + 3) * 136 + c] = vg.w;
  }
  if (tid < 128) {
    lnwS[tid] = lnw[tid];
    lnbS[tid] = lnb[tid];
    bzS[tid]  = bz[tid];
    bgS[tid]  = bg[tid];
  }
  __syncthreads();

  const v8f vzero = {0.f, 0.f, 0.f, 0.f, 0.f, 0.f, 0.f, 0.f};

  for (int tile = blockIdx.x; tile < NPIX / 64; tile += gridDim.x) {
    const int m0 = tile * 64;
    __syncthreads();   // protect LDS tiles from previous iteration

    // p is stored [d][pixel]: load 128 x 64 and transpose into tP[pix][d]
    for (int e = 0; e < 8; ++e) {
      const int idx = e * 256 + tid;   // 0..2047 float4 slots
      const int d   = idx >> 4;        // 0..127
      const int m4  = (idx & 15) * 4;  // pixel offset 0..60
      const float4 v = *(const float4*)(p + (size_t)d * NPIX + m0 + m4);
      tP[(m4 + 0) * 132 + d] = v.x;
      tP[(m4 + 1) * 132 + d] = v.y;
      tP[(m4 + 2) * 132 + d] = v.z;
      tP[(m4 + 3) * 132 + d] = v.w;
    }
    // z_norm is [pixel][c]: straight copy
    for (int e = 0; e < 8; ++e) {
      const int idx = e * 256 + tid;
      const int r   = idx >> 5;        // 0..63
      const int c4  = (idx & 31) * 4;
      const float4 v = *(const float4*)(zn + (size_t)(m0 + r) * 128 + c4);
      tZ[r * 132 + c4 + 0] = v.x;
      tZ[r * 132 + c4 + 1] = v.y;
      tZ[r * 132 + c4 + 2] = v.z;
      tZ[r * 132 + c4 + 3] = v.w;
    }
    __syncthreads();

    // LayerNorm stats: 4 threads per row, 32 elements each
    {
      const int r = tid >> 2, part = tid & 3;
      float s = 0.f, q = 0.f;
      for (int d = part * 32; d < part * 32 + 32; ++d) {
        const float v = tP[r * 132 + d];
        s += v;
        q += v * v;
      }
      redS[tid] = s;
      redQ[tid] = q;
    }
    __syncthreads();
    if ((tid & 3) == 0) {
      const int r = tid >> 2;
      const float s = redS[tid] + redS[tid + 1] + redS[tid + 2] + redS[tid + 3];
      const float q = redQ[tid] + redQ[tid + 1] + redQ[tid + 2] + redQ[tid + 3];
      const float mu  = s * (1.0f / 128.0f);
      const float var = q * (1.0f / 128.0f) - mu * mu;
      muA[r] = mu;
      rsA[r] = rsqrtf(var + 1e-5f);
    }
    __syncthreads();
    // normalize in place (becomes A matrix for the Z GEMM)
    for (int e = 0; e < 32; ++e) {
      const int idx = e * 256 + tid;   // 0..8191
      const int r   = idx >> 7;
      const int d   = idx & 127;
      tP[r * 132 + d] = (tP[r * 132 + d] - muA[r]) * rsA[r] * lnwS[d] + lnbS[d];
    }
    __syncthreads();

    // GEMM 1: Z = pn * Wz^T   (per wave: 16 rows x 64 cols, K = 128)
    v8f zacc[4], gacc[4];
    for (int ct = 0; ct < 4; ++ct) { zacc[ct] = vzero; gacc[ct] = vzero; }

    const int rowA = wrow * 16 + lrow;
    for (int k0 = 0; k0 < 128; k0 += 4) {
      const int kk = k0 + 2 * lhalf;
      v2f af;
      af.x = tP[rowA * 132 + kk];
      af.y = tP[rowA * 132 + kk + 1];
      for (int ct = 0; ct < 4; ++ct) {
        const int col = wch * 64 + ct * 16 + lrow;
        v2f bf;
        bf.x = BzT[kk * 136 + col];
        bf.y = BzT[(kk + 1) * 136 + col];
        zacc[ct] = __builtin_amdgcn_wmma_f32_16x16x4_f32(
            false, af, false, bf, (short)0, zacc[ct], false, false);
      }
    }
    // GEMM 2: G = z_norm * Wg^T
    for (int k0 = 0; k0 < 128; k0 += 4) {
      const int kk = k0 + 2 * lhalf;
      v2f af;
      af.x = tZ[rowA * 132 + kk];
      af.y = tZ[rowA * 132 + kk + 1];
      for (int ct = 0; ct < 4; ++ct) {
        const int col = wch * 64 + ct * 16 + lrow;
        v2f bf;
        bf.x = BgT[kk * 136 + col];
        bf.y = BgT[(kk + 1) * 136 + col];
        gacc[ct] = __builtin_amdgcn_wmma_f32_16x16x4_f32(
            false, af, false, bf, (short)0, gacc[ct], false, false);
      }
    }

    // epilogue: bias + sigmoid gate, write [pixel][c]
    for (int ct = 0; ct < 4; ++ct) {
      const int colg = wch * 64 + ct * 16 + lrow;
      for (int v = 0; v < 8; ++v) {
        const int row = wrow * 16 + v + 8 * lhalf;
        const size_t m = (size_t)(m0 + row);
        const float zv = zacc[ct][v] + bzS[colg];
        const float gv = gacc[ct][v] + bgS[colg];
        out[m * 128 + colg] = zv * (1.0f / (1.0f + __expf(-gv)));
      }
    }
  }
}

// ---------------------------------------------------------------------------
extern "C" void kernel_launch(void* const* d_in, const int* in_sizes, int n_in,
                              void* d_out, int out_size, void* d_ws, size_t ws_size,
                              hipStream_t stream) {
  const float* a   = (const float*)d_in[0];
  const float* b   = (const float*)d_in[1];
  const float* zn  = (const float*)d_in[2];
  const float* lnw = (const float*)d_in[3];
  const float* lnb = (const float*)d_in[4];
  const float* Wz  = (const float*)d_in[5];
  const float* bz  = (const float*)d_in[6];
  const float* Wg  = (const float*)d_in[7];
  const float* bg  = (const float*)d_in[8];
  float* out = (float*)d_out;
  float* pws = (float*)d_ws;   // 128*512*512 f32 = 134 MB staging for p ([d][i][j])

  dim3 g1(NN / 128, NN / 128, DD);
  k_bmm<<<g1, 256, 0, stream>>>(a, b, pws);

  const size_t shmem = (size_t)(2 * 128 * 136 + 2 * 64 * 132) * sizeof(float);
  k_fused<<<512, 256, shmem, stream>>>(pws, zn, lnw, lnb, Wz, bz, Wg, bg, out);
}